// ATNLPmodel_13048110645673
// MI455X (gfx1250) — compile-verified
//
#include <hip/hip_runtime.h>
#include <hip/hip_bf16.h>
#include <math.h>

// Problem constants (from the reference)
#define BQ          16          // number of queries
#define CD          1024        // feature dim C*L2
#define NDB         400000      // database rows
#define NT          (NDB / 16)  // 25000 tiles of 16 db rows
#define EPSV        1e-8f

typedef float v2f __attribute__((ext_vector_type(2)));
typedef float v8f __attribute__((ext_vector_type(8)));
typedef int   v4i __attribute__((ext_vector_type(4)));

// ---------------------------------------------------------------------------
// Kernel 1: per-query inverse norms: invq[b] = 1/(||q_b|| + eps)
// ---------------------------------------------------------------------------
__global__ __launch_bounds__(512) void qnorm_kernel(const float* __restrict__ q,
                                                    float* __restrict__ invq) {
    const int w    = threadIdx.x >> 5;   // query row
    const int lane = threadIdx.x & 31;
    float s = 0.0f;
    const float* row = q + (size_t)w * CD;
    for (int j = lane; j < CD; j += 32) {
        float v = row[j];
        s += v * v;
    }
    #pragma unroll
    for (int off = 16; off > 0; off >>= 1)
        s += __shfl_xor(s, off, 32);
    if (lane == 0)
        invq[w] = 1.0f / (sqrtf(s) + EPSV);
}

// ---------------------------------------------------------------------------
// Kernel 2: fused DB-stream GEMM (WMMA f32 16x16x4) + row-norm + tile reduce.
// Block = 256 threads = 8 waves; queries (64 KB) staged in LDS once per block
// (async global->LDS DMA when available). Each wave grid-strides over 16-row
// DB tiles; per K-step each lane issues one global_load_b128 (DB) + one
// ds_load_b128 (queries) feeding two f32 WMMA ops. DB row norms accumulate as
// packed v_pk_fma_f32 on the live b128 registers (no repacking moves).
// Partials layout: [query][tile] for a coalesced finalize scan.
// ---------------------------------------------------------------------------
__global__ __launch_bounds__(256) void knn_tile_kernel(const float* __restrict__ db,
                                                       const float* __restrict__ queries,
                                                       float* __restrict__ pmax,
                                                       int*   __restrict__ pidx,
                                                       float* __restrict__ psum) {
    __shared__ float qlds[BQ * CD];   // 64 KB query tile, reused across K loop

    const int tid  = threadIdx.x;
    const int wave = tid >> 5;
    const int lane = tid & 31;
    const int lr   = lane & 15;       // A: db row in tile; B: query column
    const int hi   = lane >> 4;       // lane half selects K sub-lanes

    // Stage all queries into LDS (async DMA path if the builtin exists).
#if __has_builtin(__builtin_amdgcn_global_load_async_to_lds_b128) && \
    __has_builtin(__builtin_amdgcn_s_wait_asynccnt)
    {
        typedef __attribute__((address_space(1))) v4i gv4i;  // global int4*
        typedef __attribute__((address_space(3))) v4i lv4i;  // LDS int4*
        gv4i* src = (gv4i*)(queries);
        lv4i* dst = (lv4i*)(qlds);
        for (int i = tid; i < (BQ * CD) / 4; i += blockDim.x) {
            __builtin_amdgcn_global_load_async_to_lds_b128(src + i, dst + i, 0, 0);
        }
        __builtin_amdgcn_s_wait_asynccnt(0);
    }
#else
    {
        const float4* src = reinterpret_cast<const float4*>(queries);
        float4*       dst = reinterpret_cast<float4*>(qlds);
        for (int i = tid; i < (BQ * CD) / 4; i += blockDim.x)
            dst[i] = src[i];
    }
#endif
    __syncthreads();

    const int wavesTotal = gridDim.x * (blockDim.x >> 5);
    const int gwave      = blockIdx.x * (blockDim.x >> 5) + wave;

    for (int tile = gwave; tile < NT; tile += wavesTotal) {
        // Per-lane base pointers: row lr of the tile, columns 4*hi + k.
        const float* arow = db + (size_t)tile * 16 * CD + (size_t)lr * CD + 4 * hi;
        const float* brow = qlds + lr * CD + 4 * hi;

        v8f acc  = {0.f, 0.f, 0.f, 0.f, 0.f, 0.f, 0.f, 0.f};
        v2f sq01 = {0.f, 0.f};   // packed sumsq accumulators -> v_pk_fma_f32
        v2f sq23 = {0.f, 0.f};

        for (int k0 = 0; k0 < CD; k0 += 32) {        // 128 B per row per outer step
            // Speculative prefetch of this row's line 1 KB ahead.
            __builtin_prefetch(arow + k0 + 256, 0, 1);
            #pragma unroll
            for (int j = 0; j < 32; j += 8) {        // K-span of 8 per b128 pair
                float4 a = *reinterpret_cast<const float4*>(arow + k0 + j);
                float4 b = *reinterpret_cast<const float4*>(brow + k0 + j);
                const v2f axy = {a.x, a.y};
                const v2f azw = {a.z, a.w};
                const v2f bxy = {b.x, b.y};
                const v2f bzw = {b.z, b.w};
                sq01 = axy * axy + sq01;             // v_pk_fma_f32, in-place regs
                sq23 = azw * azw + sq23;
                // Two 16x16x4 fp32 WMMA ops; lo lanes carry K = {k,k+1}|{k+2,k+3},
                // hi lanes carry K = {k+4,k+5}|{k+6,k+7}; A and B agree on order.
                acc = __builtin_amdgcn_wmma_f32_16x16x4_f32(
                          false, axy, false, bxy, (short)0, acc, false, false);
                acc = __builtin_amdgcn_wmma_f32_16x16x4_f32(
                          false, azw, false, bzw, (short)0, acc, false, false);
            }
        }

        // Horizontal fold (once per tile), then merge the two lane halves:
        // lanes lr and lr+16 cover disjoint K subsets of db row lr.
        float sq = (sq01.x + sq01.y) + (sq23.x + sq23.y);
        sq += __shfl_xor(sq, 16, 32);
        const float invd = 1.0f / (sqrtf(sq) + EPSV);   // lane lr holds row lr

        // Column-wise (per-query) partial reduction over the 16 db rows.
        // D layout: VGPR r, lanes 0-15 -> M=r, lanes 16-31 -> M=r+8; N = lane&15.
        float cmax = -INFINITY;
        int   cidx = 0;
        float csum = 0.0f;
        #pragma unroll
        for (int r = 0; r < 8; ++r) {
            const int   m = r + 8 * hi;                 // db row within tile
            const float s = acc[r] * __shfl(invd, m, 32);
            csum += s;
            if (s > cmax) { cmax = s; cidx = tile * 16 + m; }
        }
        // Merge the two lane halves (rows 0-7 vs 8-15 of the same column).
        {
            const float om = __shfl_xor(cmax, 16, 32);
            const int   oi = __shfl_xor(cidx, 16, 32);
            if (om > cmax) { cmax = om; cidx = oi; }
            csum += __shfl_xor(csum, 16, 32);
        }
        if (hi == 0) {
            const int o = lr * NT + tile;   // [query][tile] for coalesced scan
            pmax[o] = cmax;
            pidx[o] = cidx;
            psum[o] = csum;
        }
    }
}

// ---------------------------------------------------------------------------
// Kernel 3: final reduction over 25000 tiles per query + class lookup +
// accuracy. One block, 16 waves (one per query); contiguous per-wave scans.
// Outputs (float): [0:16) unit_sim, [16:32) avg_sim, [32:48) top_cls, [48] acc.
// ---------------------------------------------------------------------------
__global__ __launch_bounds__(512) void finalize_kernel(const float* __restrict__ pmax,
                                                       const int*   __restrict__ pidx,
                                                       const float* __restrict__ psum,
                                                       const float* __restrict__ invq,
                                                       const int*   __restrict__ db_classes,
                                                       const int*   __restrict__ y,
                                                       float* __restrict__ out) {
    __shared__ int cls_sh[BQ];
    const int tid  = threadIdx.x;
    const int w    = tid >> 5;     // query
    const int lane = tid & 31;

    const float* pm = pmax + (size_t)w * NT;
    const int*   pi = pidx + (size_t)w * NT;
    const float* ps = psum + (size_t)w * NT;

    float best = -INFINITY;
    int   bidx = 0;
    float sum  = 0.0f;
    for (int t = lane; t < NT; t += 32) {
        const float v = pm[t];
        sum += ps[t];
        if (v > best) { best = v; bidx = pi[t]; }
    }
    #pragma unroll
    for (int off = 16; off > 0; off >>= 1) {
        const float ov = __shfl_xor(best, off, 32);
        const int   oi = __shfl_xor(bidx, off, 32);
        const float os = __shfl_xor(sum,  off, 32);
        if (ov > best) { best = ov; bidx = oi; }
        sum += os;
    }
    if (lane == 0) {
        const float iq = invq[w];
        out[w]        = best * iq;                       // unit_sim
        out[16 + w]   = sum * iq * (1.0f / (float)NDB);  // avg_sim
        const int cls = db_classes[bidx];
        cls_sh[w]     = cls;
        out[32 + w]   = (float)cls;                      // top_cls (as float)
    }
    __syncthreads();
    if (tid == 0) {
        int correct = 0;
        #pragma unroll
        for (int i = 0; i < BQ; ++i) correct += (cls_sh[i] == y[i]) ? 1 : 0;
        out[48] = (float)correct / (float)BQ;            // accuracy
    }
}

// ---------------------------------------------------------------------------
// Host-side launch
// ---------------------------------------------------------------------------
extern "C" void kernel_launch(void* const* d_in, const int* in_sizes, int n_in,
                              void* d_out, int out_size, void* d_ws, size_t ws_size,
                              hipStream_t stream) {
    const float* queries    = (const float*)d_in[0];   // [16, 32, 32] fp32
    const float* database   = (const float*)d_in[1];   // [400000, 32, 32] fp32
    const int*   db_classes = (const int*)d_in[2];     // [400000] int32
    const int*   y          = (const int*)d_in[3];     // [16] int32
    float*       out        = (float*)d_out;           // 49 floats

    // Workspace carve-up: invq[16] | pmax[16*NT] | psum[16*NT] | pidx[16*NT]
    float* invq = (float*)d_ws;
    float* pmax = invq + 16;
    float* psum = pmax + (size_t)BQ * NT;
    int*   pidx = (int*)(psum + (size_t)BQ * NT);
    (void)in_sizes; (void)n_in; (void)out_size; (void)ws_size;

    qnorm_kernel<<<1, 512, 0, stream>>>(queries, invq);
    knn_tile_kernel<<<1024, 256, 0, stream>>>(database, queries, pmax, pidx, psum);
    finalize_kernel<<<1, 512, 0, stream>>>(pmax, pidx, psum, invq, db_classes, y, out);
}